// CGNNLayer_18932215841129
// MI455X (gfx1250) — compile-verified
//
#include <hip/hip_runtime.h>
#include <math.h>

#define B_ 4
#define N_ 256
#define H_ 128
#define R_ 32
#define IN_ 544  // 2H + R + 2H

typedef float v2f __attribute__((ext_vector_type(2)));
typedef float v8f __attribute__((ext_vector_type(8)));

__device__ __forceinline__ float silu_f(float v) {
    // v * sigmoid(v) with fast (~1 ulp) v_rcp_f32 instead of IEEE divide chain
    return v * __builtin_amdgcn_rcpf(1.0f + __expf(-v));
}

__device__ __forceinline__ void wait_async0() {
#if defined(__has_builtin) && __has_builtin(__builtin_amdgcn_s_wait_asynccnt)
    __builtin_amdgcn_s_wait_asynccnt(0);
#else
    asm volatile("s_wait_asynccnt 0x0" ::: "memory");
#endif
}

// ---------------------------------------------------------------------------
// Kernel 1: per-node precompute.
//   base[b,i,h] = x[b,i,:].W1a[h,:] + cond[b,:].W1d[h,:] + b1[h]
//   xb  [b,j,h] = x[b,j,:].W1b[h,:]
// W1 row layout (544): [0:128]=x_i, [128:256]=x_j, [256:288]=adj, [288:544]=cond
// ---------------------------------------------------------------------------
__global__ void k_precompute(const float* __restrict__ x,
                             const float* __restrict__ cond,
                             const float* __restrict__ W1,
                             const float* __restrict__ b1,
                             float* __restrict__ base,
                             float* __restrict__ xb) {
    int bi = blockIdx.x;          // b*N + i
    int b  = bi / N_;
    int h  = threadIdx.x;         // 0..127

    __shared__ __align__(16) float sx[H_];
    __shared__ __align__(16) float sc[2 * H_];
    sx[h]      = x[bi * H_ + h];
    sc[h]      = cond[b * 2 * H_ + h];
    sc[H_ + h] = cond[b * 2 * H_ + H_ + h];
    __syncthreads();

    const float4* wa = (const float4*)(W1 + (size_t)h * IN_);
    const float4* wb = (const float4*)(W1 + (size_t)h * IN_ + H_);
    const float4* wd = (const float4*)(W1 + (size_t)h * IN_ + 2*H_ + R_);
    const float4* xv = (const float4*)sx;
    const float4* cv = (const float4*)sc;

    float a0 = b1[h];
    float a1 = 0.0f;
#pragma unroll 8
    for (int d = 0; d < H_ / 4; ++d) {
        float4 va = wa[d], vb = wb[d], v = xv[d];
        a0 += va.x * v.x + va.y * v.y + va.z * v.z + va.w * v.w;
        a1 += vb.x * v.x + vb.y * v.y + vb.z * v.z + vb.w * v.w;
    }
#pragma unroll 8
    for (int d = 0; d < 2 * H_ / 4; ++d) {
        float4 vd = wd[d], v = cv[d];
        a0 += vd.x * v.x + vd.y * v.y + vd.z * v.z + vd.w * v.w;
    }
    base[bi * H_ + h] = a0;
    xb[bi * H_ + h]   = a1;
}

// ---------------------------------------------------------------------------
// Kernel 2 (WMMA + async-to-LDS): per-edge adj GEMM + silu + masked aggregation.
//   acc[b,i,h] = sum_j mask[b,j] * silu(base[b,i,h] + xb[b,j,h]
//                                       + adj[b,i,j,:].W1c[h,:])
// One block per (b,i); 8 waves; wave w owns channels [16w,16w+16).
// Per j-step: 16x16 tile via 8x V_WMMA_F32_16X16X4_F32 (K=32), with adj tiles
// double-buffered in LDS via global_load_async_to_lds_b64 (ASYNCcnt).
// A (16x4 f32): lane l holds M=l%16, K = comp + 2*(l/16).
// C (16x16 f32): lane l holds N=l%16, M = vgpr + 8*(l/16).
// B (4x16 f32) assumed mirror of A: lane l holds N=l%16, K = comp + 2*(l/16).
// ---------------------------------------------------------------------------
__global__ void k_edges(const float* __restrict__ adj,
                        const float* __restrict__ base,
                        const float* __restrict__ xb,
                        const int*   __restrict__ mask,
                        const float* __restrict__ W1,
                        float* __restrict__ acc_out) {
    int bi   = blockIdx.x;            // b*N + i
    int b    = bi / N_;
    int tid  = threadIdx.x;           // 0..255 (8 waves)
    int w    = tid >> 5;
    int lane = tid & 31;
    int half = lane >> 4;
    int lp   = lane & 15;
    int ch   = w * 16 + lp;

    __shared__ __align__(16) float sAdj[2][16 * R_];  // 2 x 2KB double buffer
    __shared__ float sMask[N_];

    sMask[tid] = (float)mask[b * N_ + tid];           // covered by first barrier

    const float* adjRow = adj + (size_t)bi * N_ * R_;   // adj[b,i,:,:]

    // LDS byte addresses for this thread's async destinations (ISA: generic
    // LDS-aperture address low 32 bits == workgroup-relative LDS address).
    unsigned ldsDst[2];
    ldsDst[0] = (unsigned)(size_t)&sAdj[0][tid * 2];
    ldsDst[1] = (unsigned)(size_t)&sAdj[1][tid * 2];

    // Issue async stage of tile 0 into buffer 0 (8 bytes/thread = 2KB tile).
    {
        unsigned goff = (unsigned)(tid * 8);
        asm volatile("global_load_async_to_lds_b64 %0, %1, %2"
                     :: "v"(ldsDst[0]), "v"(goff), "s"(adjRow) : "memory");
    }

    // B fragments (constant): B(k,n) = W1c[n,k] = W1[n*544 + 256 + k]
    v2f bfrag[8];
    const float* wrow = W1 + (size_t)ch * IN_ + 2 * H_;
#pragma unroll
    for (int kk = 0; kk < 8; ++kk) {
        const float* p = wrow + kk * 4 + 2 * half;
        v2f t; t.x = p[0]; t.y = p[1];
        bfrag[kk] = t;
    }
    float basec = base[bi * H_ + ch];
    float acc   = 0.0f;

    wait_async0();
    __syncthreads();

    for (int js = 0; js < N_ / 16; ++js) {
        int buf = js & 1;
        // Stage next tile into the other buffer while we compute this one.
        if (js + 1 < N_ / 16) {
            unsigned goff = (unsigned)((js + 1) * 16 * R_ * 4 + tid * 8);
            asm volatile("global_load_async_to_lds_b64 %0, %1, %2"
                         :: "v"(ldsDst[buf ^ 1]), "v"(goff), "s"(adjRow)
                         : "memory");
        }
        if (js + 2 < N_ / 16)
            __builtin_prefetch(adjRow + (js + 2) * 16 * R_ + tid * 2, 0, 3);

        v8f c = {0.f, 0.f, 0.f, 0.f, 0.f, 0.f, 0.f, 0.f};
#pragma unroll
        for (int kk = 0; kk < 8; ++kk) {
            v2f a = *(const v2f*)&sAdj[buf][lp * R_ + kk * 4 + 2 * half];
            c = __builtin_amdgcn_wmma_f32_16x16x4_f32(
                    /*neg_a=*/false, a, /*neg_b=*/false, bfrag[kk],
                    /*c_mod=*/(short)0, c, /*reuse_a=*/false, /*reuse_b=*/false);
        }

        // Epilogue: tile row M=v+8*half is sender j = js*16 + M.
        float part = 0.0f;
#pragma unroll
        for (int v = 0; v < 8; ++v) {
            int j = js * 16 + v + 8 * half;
            float pre = c[v] + basec + xb[((size_t)b * N_ + j) * H_ + ch];
            part += sMask[j] * silu_f(pre);
        }
        acc += part;

        wait_async0();      // next tile fully in LDS
        __syncthreads();    // all waves done reading current buffer
    }

    // Combine the two lane halves (same channel, disjoint j sets).
    acc += __shfl_xor(acc, 16, 32);
    if (half == 0)
        acc_out[bi * H_ + ch] = acc;
}

// ---------------------------------------------------------------------------
// Kernel 3: node update + residual + LayerNorm.
// ---------------------------------------------------------------------------
__global__ void k_update(const float* __restrict__ x,
                         const float* __restrict__ acc,
                         const int*   __restrict__ mask,
                         const float* __restrict__ W2, const float* __restrict__ b2,
                         const float* __restrict__ W3, const float* __restrict__ b3,
                         const float* __restrict__ W4, const float* __restrict__ b4,
                         const float* __restrict__ gamma, const float* __restrict__ beta,
                         float* __restrict__ out) {
    int bi = blockIdx.x;
    int b  = bi / N_;
    int h  = threadIdx.x;

    __shared__ __align__(16) float sX[H_];
    __shared__ __align__(16) float sA[H_];
    __shared__ __align__(16) float sAg[H_];
    __shared__ __align__(16) float sU[H_];
    __shared__ float sR[H_];

    sX[h] = x[bi * H_ + h];
    sA[h] = acc[bi * H_ + h];
    sR[h] = (float)(mask[b * N_ + h] + mask[b * N_ + H_ + h]);
    __syncthreads();

    for (int s = 64; s > 0; s >>= 1) {
        if (h < s) sR[h] += sR[h + s];
        __syncthreads();
    }
    float cnt = sR[0];
    __syncthreads();

    float ag = b2[h] * cnt;
    {
        const float4* wr = (const float4*)(W2 + (size_t)h * H_);
        const float4* av = (const float4*)sA;
#pragma unroll 8
        for (int d = 0; d < H_ / 4; ++d) {
            float4 a = wr[d], v = av[d];
            ag += a.x * v.x + a.y * v.y + a.z * v.z + a.w * v.w;
        }
    }
    sAg[h] = ag;
    __syncthreads();

    float ua = b3[h];
    {
        const float4* w0 = (const float4*)(W3 + (size_t)h * 2 * H_);
        const float4* w1 = (const float4*)(W3 + (size_t)h * 2 * H_ + H_);
        const float4* xv = (const float4*)sX;
        const float4* gv = (const float4*)sAg;
#pragma unroll 8
        for (int d = 0; d < H_ / 4; ++d) {
            float4 a = w0[d], v = xv[d];
            ua += a.x * v.x + a.y * v.y + a.z * v.z + a.w * v.w;
        }
#pragma unroll 8
        for (int d = 0; d < H_ / 4; ++d) {
            float4 a = w1[d], v = gv[d];
            ua += a.x * v.x + a.y * v.y + a.z * v.z + a.w * v.w;
        }
    }
    sU[h] = silu_f(ua);
    __syncthreads();

    float up = b4[h];
    {
        const float4* wr = (const float4*)(W4 + (size_t)h * H_);
        const float4* uv = (const float4*)sU;
#pragma unroll 8
        for (int d = 0; d < H_ / 4; ++d) {
            float4 a = wr[d], v = uv[d];
            up += a.x * v.x + a.y * v.y + a.z * v.z + a.w * v.w;
        }
    }
    float y = sX[h] + up;

    sR[h] = y;
    __syncthreads();
    for (int s = 64; s > 0; s >>= 1) {
        if (h < s) sR[h] += sR[h + s];
        __syncthreads();
    }
    float mu = sR[0] * (1.0f / H_);
    __syncthreads();
    float dv = y - mu;
    sR[h] = dv * dv;
    __syncthreads();
    for (int s = 64; s > 0; s >>= 1) {
        if (h < s) sR[h] += sR[h + s];
        __syncthreads();
    }
    float var = sR[0] * (1.0f / H_);
    out[bi * H_ + h] = dv * rsqrtf(var + 1e-5f) * gamma[h] + beta[h];
}

// ---------------------------------------------------------------------------
extern "C" void kernel_launch(void* const* d_in, const int* in_sizes, int n_in,
                              void* d_out, int out_size, void* d_ws, size_t ws_size,
                              hipStream_t stream) {
    (void)in_sizes; (void)n_in; (void)out_size; (void)ws_size;

    const float* x     = (const float*)d_in[0];
    const float* adj   = (const float*)d_in[1];
    const int*   mask  = (const int*)  d_in[2];
    const float* cond  = (const float*)d_in[3];
    const float* W1    = (const float*)d_in[4];
    const float* b1    = (const float*)d_in[5];
    const float* W2    = (const float*)d_in[6];
    const float* b2    = (const float*)d_in[7];
    const float* W3    = (const float*)d_in[8];
    const float* b3    = (const float*)d_in[9];
    const float* W4    = (const float*)d_in[10];
    const float* b4    = (const float*)d_in[11];
    const float* gamma = (const float*)d_in[12];
    const float* beta  = (const float*)d_in[13];
    float* out = (float*)d_out;

    const size_t nodes = (size_t)B_ * N_ * H_;   // 131072 floats
    float* base = (float*)d_ws;
    float* xb   = base + nodes;
    float* accb = xb + nodes;

    k_precompute<<<B_ * N_, H_,  0, stream>>>(x, cond, W1, b1, base, xb);
    k_edges     <<<B_ * N_, 256, 0, stream>>>(adj, base, xb, mask, W1, accb);
    k_update    <<<B_ * N_, H_,  0, stream>>>(x, accb, mask, W2, b2, W3, b3,
                                              W4, b4, gamma, beta, out);
}